// ChamferLoss_89532888252875
// MI455X (gfx1250) — compile-verified
//
#include <hip/hip_runtime.h>

typedef float v2f __attribute__((ext_vector_type(2)));
typedef float v8f __attribute__((ext_vector_type(8)));

#define NPTS   8192
#define NBATCH 4
#define TILES  (NPTS / 16)
#define WAVES_PER_WG 8
#define ITERS  (TILES / (2 * WAVES_PER_WG))   // 32

// Single v_min_num_f32 without IEEE canonicalization traffic.
__device__ __forceinline__ float min_num(float a, float b) {
    float r;
    asm("v_min_num_f32 %0, %1, %2" : "=v"(r) : "v"(a), "v"(b));
    return r;
}

// Cross-lane XOR swizzle (group-of-32 mode: and=0x1f, or=0, xor=MASK).
template <int MASK>
__device__ __forceinline__ float swz_xor(float v) {
    return __int_as_float(
        __builtin_amdgcn_ds_swizzle(__float_as_int(v), 0x1f | (MASK << 10)));
}

// Build fragment-ready forms of a point set:
//   Aform[i] = (-2x, -2y, -2z, 1)    (query rows)
//   Bform[i] = ( x,   y,   z, |p|^2) (target cols; .w also = deferred |q|^2)
__global__ __launch_bounds__(256)
void chamfer_prep_kernel(const float* __restrict__ pts,
                         float* __restrict__ Aform,
                         float* __restrict__ Bform, int total)
{
    const int i = blockIdx.x * 256 + threadIdx.x;
    if (i >= total) return;
    const float x = pts[i * 3 + 0];
    const float y = pts[i * 3 + 1];
    const float z = pts[i * 3 + 2];
    const float n = x * x + y * y + z * z;
    float4 av; av.x = -2.0f * x; av.y = -2.0f * y; av.z = -2.0f * z; av.w = 1.0f;
    float4 bv; bv.x = x;         bv.y = y;         bv.z = z;         bv.w = n;
    ((float4*)Aform)[i] = av;
    ((float4*)Bform)[i] = bv;
}

// One block = 8 wave32s. Block owns one 16-row query tile (blockIdx.x) of
// batch blockIdx.y; waves stripe the 512 target tiles, 2 tiles/iteration,
// compile-time trip count (uniform loop, no exec masking).
//   D[n][m] = A x B = -2 q.p + |p|^2 ;  min_m D + |q|^2 = min_m dist^2.
// f32 16x4 fragment layout (wave32): lanes 0-15 hold K=0,1; lanes 16-31 hold
// K=2,3 -> each lane's fragment is one aligned b64 at point*16 + hi*8.
// C/D: lane%16 = column, VGPR i = row i (lanes<16) / row i+8 (lanes>=16).
__global__ __launch_bounds__(256)
void chamfer_nn_min_kernel(const float* __restrict__ qA,   // [B*NPTS*4] query A-form
                           const float* __restrict__ tB,   // [B*NPTS*4] target B-form
                           const float* __restrict__ qB,   // [B*NPTS*4] query B-form (.w)
                           float* __restrict__ minbuf)     // [B*NPTS]
{
    const int lane = threadIdx.x & 31;
    const int wave = threadIdx.x >> 5;
    const int col  = lane & 15;   // column index within 16-wide tile
    const int hi   = lane >> 4;   // K-split half of the wave
    const int b    = blockIdx.y;
    const int qbase = blockIdx.x * 16;

    const float* qAb = qA + (size_t)b * NPTS * 4;
    const float* tBb = tB + (size_t)b * NPTS * 4;

    // A fragment: one aligned b64 per lane, loop-invariant.
    const v2f a = *(const v2f*)(qAb + (qbase + col) * 4 + hi * 2);

    v8f rm0, rm1;
    #pragma unroll
    for (int i = 0; i < 8; ++i) { rm0[i] = 3.0e38f; rm1[i] = 3.0e38f; }

    // Lane's pointer into the target B-form stream for this wave's tiles.
    const float* pB = tBb + wave * 128 + col * 4 + hi * 2;

    #pragma unroll 2
    for (int it = 0; it < ITERS; ++it) {
        const v2f b0 = *(const v2f*)(pB);        // tile 2*wave + 16*it
        const v2f b1 = *(const v2f*)(pB + 64);   // tile 2*wave + 16*it + 1
        pB += 2 * WAVES_PER_WG * 16 * 4;

        v8f c0 = {};
        c0 = __builtin_amdgcn_wmma_f32_16x16x4_f32(
                 false, a, false, b0, (short)0, c0, false, false);
        #pragma unroll
        for (int i = 0; i < 8; ++i) rm0[i] = min_num(rm0[i], c0[i]);

        v8f c1 = {};
        c1 = __builtin_amdgcn_wmma_f32_16x16x4_f32(
                 false, a, false, b1, (short)0, c1, false, false);
        #pragma unroll
        for (int i = 0; i < 8; ++i) rm1[i] = min_num(rm1[i], c1[i]);
    }

    #pragma unroll
    for (int i = 0; i < 8; ++i) rm0[i] = min_num(rm0[i], rm1[i]);

    // Min over the 16 columns: XOR-swizzle reduce within each 16-lane half.
    #pragma unroll
    for (int i = 0; i < 8; ++i) {
        float v = rm0[i];
        v = min_num(v, swz_xor<1>(v));
        v = min_num(v, swz_xor<2>(v));
        v = min_num(v, swz_xor<4>(v));
        v = min_num(v, swz_xor<8>(v));
        rm0[i] = v;
    }

    // Combine the 8 waves' partial row-mins through LDS.
    __shared__ float smin[WAVES_PER_WG][16];
    if (col == 0) {
        #pragma unroll
        for (int i = 0; i < 8; ++i)
            smin[wave][hi * 8 + i] = rm0[i];   // hi=0: rows 0-7, hi=1: rows 8-15
    }
    __syncthreads();
    if (threadIdx.x < 16) {
        const int t = threadIdx.x;
        float v = smin[0][t];
        #pragma unroll
        for (int w = 1; w < WAVES_PER_WG; ++w) v = min_num(v, smin[w][t]);
        // Re-apply the deferred row constant |q|^2 (query B-form .w).
        const float qn = qB[((size_t)b * NPTS + qbase + t) * 4 + 3];
        minbuf[(size_t)b * NPTS + qbase + t] = v + qn;
    }
}

// Single-block sum of both min buffers (contiguous), scaled by 1/(B*N).
__global__ __launch_bounds__(256)
void chamfer_finalize_kernel(const float* __restrict__ minbuf,
                             float* __restrict__ out, int total, float scale)
{
    __shared__ float s[256];
    float acc = 0.0f;
    for (int i = threadIdx.x; i < total; i += 256) acc += minbuf[i];
    s[threadIdx.x] = acc;
    __syncthreads();
    for (int off = 128; off > 0; off >>= 1) {
        if ((int)threadIdx.x < off) s[threadIdx.x] += s[threadIdx.x + off];
        __syncthreads();
    }
    if (threadIdx.x == 0) out[0] = s[0] * scale;
}

extern "C" void kernel_launch(void* const* d_in, const int* in_sizes, int n_in,
                              void* d_out, int out_size, void* d_ws, size_t ws_size,
                              hipStream_t stream) {
    const float* pred = (const float*)d_in[0];   // [4][8192][3]
    const float* gt   = (const float*)d_in[1];   // [4][8192][3]

    const size_t NPT  = (size_t)NBATCH * NPTS;   // points per set
    const size_t SZ4  = NPT * 4;                 // floats per form array
    float* ws    = (float*)d_ws;
    float* predA = ws;                           // (-2x,-2y,-2z,1)
    float* predB = ws + SZ4;                     // ( x,  y,  z, |x|^2)
    float* gtA   = ws + 2 * SZ4;
    float* gtB   = ws + 3 * SZ4;
    float* minP  = ws + 4 * SZ4;                 // [B*NPTS] pred->gt NN mins
    float* minG  = minP + NPT;                   // [B*NPTS] gt->pred NN mins

    const int prepBlocks = (int)((NPT + 255) / 256);
    chamfer_prep_kernel<<<prepBlocks, 256, 0, stream>>>(pred, predA, predB, (int)NPT);
    chamfer_prep_kernel<<<prepBlocks, 256, 0, stream>>>(gt,   gtA,   gtB,   (int)NPT);

    dim3 grid(TILES, NBATCH);
    dim3 block(256);
    chamfer_nn_min_kernel<<<grid, block, 0, stream>>>(predA, gtB,   predB, minP);
    chamfer_nn_min_kernel<<<grid, block, 0, stream>>>(gtA,   predB, gtB,   minG);

    chamfer_finalize_kernel<<<1, 256, 0, stream>>>(
        minP, (float*)d_out, 2 * NBATCH * NPTS,
        1.0f / ((float)NBATCH * (float)NPTS));
}